// Bicubic_67284957659488
// MI455X (gfx1250) — compile-verified
//
#include <hip/hip_runtime.h>

typedef __attribute__((ext_vector_type(2))) float v2f;
typedef __attribute__((ext_vector_type(4))) float v4f;
typedef __attribute__((ext_vector_type(8))) float v8f;

#define IN_H 768
#define IN_W 768
#define OUT_H 384
#define OUT_W 384
#define NPLANES 12                          // B*C = 4*3
#define TILES_X (OUT_W / 16)                // 24
#define TILES_PER_PLANE (TILES_X * TILES_X) // 576
#define WAVES_PER_BLOCK 8

#define W0f (-0.0703125f / 2.046875f)
#define W1f ( 0.2265625f / 2.046875f)
#define W2f ( 0.8671875f / 2.046875f)

// Normalized constant 6-tap weights: cubic(0.5*(2.5-p)) / 2.046875
__device__ __forceinline__ float w1sel(int p) {
  return (p == 0 || p == 5) ? W0f : ((p == 1 || p == 4) ? W1f : W2f);
}

// Banded weight: w1[p] for 0<=p<6, else 0. Branchless (v_cndmask chain),
// avoids dynamically indexed LDS (which compiled to predicated ds_loads
// with exec-mask save/restore before every WMMA).
__device__ __forceinline__ float wband(int p) {
  float v = 0.0f;
  v = (p == 0 || p == 5) ? W0f : v;
  v = (p == 1 || p == 4) ? W1f : v;
  v = (p == 2 || p == 3) ? W2f : v;
  return v;
}

__device__ __forceinline__ int iclamp(int v, int lo, int hi) {
  return v < lo ? lo : (v > hi ? hi : v);
}

// LDS element index for T[k][n] (40x16): pairs of adjacent k are
// interleaved so each B register pair is one aligned ds_load_b64.
__device__ __forceinline__ int tidx(int k, int n) {
  return (k >> 1) * 32 + 2 * n + (k & 1);
}

// One wave32 computes one 16x16 output tile as Wv(16x40) x T(40x16) via
// 10 chained V_WMMA_F32_16X16X4_F32. T is the horizontally filtered tile
// staged in LDS; Wv is the banded constant weight matrix built per lane.
__global__ __launch_bounds__(256) void bicubic_resize_wmma(
    const float* __restrict__ in, float* __restrict__ res) {
  __shared__ float tile[WAVES_PER_BLOCK][40 * 16];
  const int tid  = threadIdx.x;
  const int lane = tid & 31;
  const int wv   = tid >> 5;

  const int T     = blockIdx.x * WAVES_PER_BLOCK + wv;  // 0..6911
  const int plane = T / TILES_PER_PLANE;                // 0..11
  const int tt    = T - plane * TILES_PER_PLANE;
  const int ty    = tt / TILES_X;
  const int tx    = tt - ty * TILES_X;
  const int OY = ty * 16, OX = tx * 16;
  const float* __restrict__ ip = in + (size_t)plane * (IN_H * IN_W);
  float* tl = tile[wv];
  const int r0 = 2 * OY - 2;           // first (unclamped) input row of band
  const int m  = lane & 15;
  const int hi = lane >> 4;

  // Horizontal pass: T[k][n] = sum_q w[q] * in[clamp(r0+k)][clamp(2*(OX+n)-2+q)]
  // rows 0..35 real; rows 36..39 zeroed (their Wv columns are zero).
#pragma unroll
  for (int i = 0; i < 20; ++i) {
    const int e = lane + 32 * i;       // 0..639
    const int k = e >> 4;
    const int n = e & 15;
    float acc = 0.0f;
    if (k < 36) {
      const int row = iclamp(r0 + k, 0, IN_H - 1);
      const float* __restrict__ rp = ip + (size_t)row * IN_W;
      const int cb = 2 * (OX + n) - 2;
#pragma unroll
      for (int q = 0; q < 6; ++q)
        acc += w1sel(q) * rp[iclamp(cb + q, 0, IN_W - 1)];
    }
    tl[tidx(k, n)] = acc;
  }
  __syncthreads();

  // Vertical pass: D = A x B + C, K=40 in 10 steps.
  // Layouts (wave32): A reg j, lane L -> M=L%16, K=2*(L>>4)+j
  //                   B reg j, lane L -> K=2*(L>>4)+j, N=L%16
  //                   C reg r, lane L -> M=r+8*(L>>4), N=L%16
  // Preload all B pairs (one b64 each) and build all A pairs (branchless
  // constants) so the 10 WMMAs issue back-to-back after one ds wait.
  v2f breg[10], areg[10];
#pragma unroll
  for (int t = 0; t < 10; ++t) {
    breg[t] = *(const v2f*)(tl + (2 * t + hi) * 32 + 2 * m); // T[4t+2hi..+1][m]
    const int k0 = 4 * t + 2 * hi;
    areg[t][0] = wband(k0 + 0 - 2 * m);
    areg[t][1] = wband(k0 + 1 - 2 * m);
  }

  v8f c = {};
#pragma unroll
  for (int t = 0; t < 10; ++t) {
    c = __builtin_amdgcn_wmma_f32_16x16x4_f32(
        /*neg_a=*/false, areg[t], /*neg_b=*/false, breg[t],
        /*c_mod=*/(short)0, c, /*reuse_a=*/false, /*reuse_b=*/false);
  }

  float* rp = res + (size_t)plane * (OUT_H * OUT_W)
                  + (size_t)(OY + 8 * hi) * OUT_W + OX + m;
#pragma unroll
  for (int r = 0; r < 8; ++r)
    __builtin_nontemporal_store(c[r], rp + (size_t)r * OUT_W);
}

// Broadcast kernel-weight output: 85 MB of the same 36 constants.
// Each thread emits one float4 non-temporal store (fully coalesced).
__global__ __launch_bounds__(256) void bicubic_fill_weights(float* __restrict__ out) {
  __shared__ float w72[72];  // doubled 36-pattern to handle float4 wraparound
  const int tid = threadIdx.x;
  if (tid < 72) {
    const int i = tid >= 36 ? tid - 36 : tid;
    w72[tid] = w1sel(i / 6) * w1sel(i - (i / 6) * 6);
  }
  __syncthreads();
  const size_t g = (size_t)blockIdx.x * 256 + tid;
  const int base = (int)((g * 4) % 36);
  v4f v;
  v.x = w72[base + 0];
  v.y = w72[base + 1];
  v.z = w72[base + 2];
  v.w = w72[base + 3];
  __builtin_nontemporal_store(v, (v4f*)out + g);
}

extern "C" void kernel_launch(void* const* d_in, const int* in_sizes, int n_in,
                              void* d_out, int out_size, void* d_ws, size_t ws_size,
                              hipStream_t stream) {
  (void)in_sizes; (void)n_in; (void)d_ws; (void)ws_size; (void)out_size;
  const float* in = (const float*)d_in[0];
  float* res = (float*)d_out;                                   // 4*3*384*384
  float* ker = (float*)d_out + (size_t)NPLANES * OUT_H * OUT_W; // 4*1*384*384*36

  const int resize_blocks = (NPLANES * TILES_PER_PLANE) / WAVES_PER_BLOCK; // 864
  bicubic_resize_wmma<<<resize_blocks, 256, 0, stream>>>(in, res);

  const size_t kerN4 = ((size_t)4 * OUT_H * OUT_W * 36) / 4;    // 5,308,416 float4s
  const int fill_blocks = (int)(kerN4 / 256);                   // 20,736
  bicubic_fill_weights<<<fill_blocks, 256, 0, stream>>>(ker);
}